// ConvAct_56796647522798
// MI455X (gfx1250) — compile-verified
//
#include <hip/hip_runtime.h>
#include <hip/hip_bf16.h>

typedef __attribute__((ext_vector_type(2))) float v2f;
typedef __attribute__((ext_vector_type(8))) float v8f;
typedef __attribute__((ext_vector_type(4))) int   v4i;

// address-space qualified v4i for the async-to-LDS builtin:
// clang HIP prints AS1 as __device__, AS3 as __shared__
typedef __attribute__((address_space(1))) v4i as1_v4i;   // global
typedef __attribute__((address_space(3))) v4i as3_v4i;   // LDS

#define DFEAT 128
#define TM 16
#define LDS_STRIDE 260   // 1040B rows: 16B-aligned for B128 async writes; banks (4m+c)%64 -> conflict-free A-frag reads

// low 32 bits of a generic pointer to __shared__ = LDS byte address
__device__ __forceinline__ unsigned lds_off32(const void* p) {
    return (unsigned)(unsigned long long)p;
}

// CDNA5 async memory->LDS copy (16B), tracked by ASYNCcnt
__device__ __forceinline__ void async_copy_b128(unsigned lds_addr, const float* g) {
#if __has_builtin(__builtin_amdgcn_global_load_async_to_lds_b128)
    __builtin_amdgcn_global_load_async_to_lds_b128(
        (as1_v4i*)(unsigned long long)(const void*)g,
        (as3_v4i*)(unsigned long long)lds_addr,
        0, 0);
#else
    asm volatile("global_load_async_to_lds_b128 %0, %1, off"
                 :: "v"(lds_addr), "v"(g) : "memory");
#endif
}

__device__ __forceinline__ void wait_asynccnt0() {
#if __has_builtin(__builtin_amdgcn_s_wait_asynccnt)
    __builtin_amdgcn_s_wait_asynccnt(0);
#else
    asm volatile("s_wait_asynccnt 0x0" ::: "memory");
#endif
}

// ---------------------------------------------------------------------------
// Kernel 1: edge scatter (mean-agg numerator + degree), one wave32 per edge.
// Random x-row gather (b128/lane) + 4 no-return global_atomic_add_f32/lane.
// x (51MB) and msg (51MB) both fit in the 192MB L2 -> atomics run at L2 rate.
// ---------------------------------------------------------------------------
__global__ __launch_bounds__(256) void sage_scatter_kernel(
    const float* __restrict__ x, const int* __restrict__ edge_index,
    float* __restrict__ msg, float* __restrict__ deg, int n_edges)
{
    long long g = (long long)blockIdx.x * blockDim.x + threadIdx.x;
    int edge = (int)(g >> 5);
    int lane = (int)(g & 31);
    if (edge >= n_edges) return;

    int src = edge_index[edge];             // row 0 of edge_index
    int dst = edge_index[n_edges + edge];   // row 1 of edge_index

    const float4* xs = reinterpret_cast<const float4*>(x + (size_t)src * DFEAT);
    float4 v = xs[lane];

    float* m = msg + (size_t)dst * DFEAT + lane * 4;
    atomicAdd(m + 0, v.x);
    atomicAdd(m + 1, v.y);
    atomicAdd(m + 2, v.z);
    atomicAdd(m + 3, v.w);
    if (lane == 0) atomicAdd(deg + dst, 1.0f);
}

// ---------------------------------------------------------------------------
// Kernel 2: fused dual-GEMM + bias + ReLU via V_WMMA_F32_16X16X4_F32.
//   out = relu( scale_m * (msg @ Wl^T) + (x @ Wr^T) + bl ),  scale_m = 1/max(deg,1)
// Block = one 16-row M tile; 8 waves, wave w owns N cols [16w, 16w+16).
// A panel (16 x 256 = [raw msg | x]) staged via GLOBAL_LOAD_ASYNC_TO_LDS_B128.
// Two accumulators keep the per-row mean scale out of the K loop (linearity).
// ---------------------------------------------------------------------------
__global__ __launch_bounds__(256) void sage_gemm_kernel(
    const float* __restrict__ x, const float* __restrict__ msg,
    const float* __restrict__ deg,
    const float* __restrict__ Wl, const float* __restrict__ bl,
    const float* __restrict__ Wr,
    float* __restrict__ out, int n_nodes)
{
    __shared__ float Atile[TM * LDS_STRIDE];
    __shared__ float scale[TM];

    const int m0  = blockIdx.x * TM;
    const int tid = threadIdx.x;

    // per-row 1/max(deg,1) (consumed only in the epilogue)
    if (tid < TM) {
        int r = m0 + tid; if (r >= n_nodes) r = n_nodes - 1;
        scale[tid] = 1.0f / fmaxf(deg[r], 1.0f);
    }

    // async stage of the 16x256 A panel: cols [0,128)=msg, [128,256)=x.
    // 1024 16B chunks; 4 per thread; LDS rows padded to 260 floats (16B aligned).
    const unsigned abase = lds_off32(Atile);
#pragma unroll
    for (int i = 0; i < 4; ++i) {
        int q  = tid + i * 256;          // chunk id 0..1023
        int r  = q >> 6;                 // row 0..15   (64 chunks per row)
        int c4 = (q & 63) << 2;          // col 0..252 step 4
        int row = m0 + r; if (row >= n_nodes) row = n_nodes - 1;
        const float* gp = (c4 < DFEAT)
            ? (msg + (size_t)row * DFEAT + c4)
            : (x   + (size_t)row * DFEAT + (c4 - DFEAT));
        async_copy_b128(abase + (unsigned)(r * LDS_STRIDE + c4) * 4u, gp);
    }
    wait_asynccnt0();
    __syncthreads();

    const int wave = tid >> 5;
    const int lane = tid & 31;
    const int n0   = wave * 16;
    const int ln   = lane & 15;          // B/C/D column within tile; A row m
    const int lk   = (lane >> 4) * 2;    // K sub-offset per ISA A/B layout

    v8f acc1, acc2;                      // acc1 = msg@Wl^T ; acc2 = x@Wr^T (+bias)
    const float bias = bl[n0 + ln];
#pragma unroll
    for (int j = 0; j < 8; ++j) { acc1[j] = 0.0f; acc2[j] = bias; }

#pragma unroll 8
    for (int kk = 0; kk < 32; ++kk) {    // K loop over msg / Wl
        const int kb = kk * 4;
        v2f a, b;
        const float* ap = &Atile[ln * LDS_STRIDE + kb + lk];
        a.x = ap[0]; a.y = ap[1];
        const float* bp = &Wl[(size_t)(n0 + ln) * DFEAT + kb + lk];
        b.x = bp[0]; b.y = bp[1];
        acc1 = __builtin_amdgcn_wmma_f32_16x16x4_f32(
            false, a, false, b, (short)0, acc1, false, false);
    }
#pragma unroll 8
    for (int kk = 0; kk < 32; ++kk) {    // K loop over x / Wr
        const int kb = kk * 4;
        v2f a, b;
        const float* ap = &Atile[ln * LDS_STRIDE + DFEAT + kb + lk];
        a.x = ap[0]; a.y = ap[1];
        const float* bp = &Wr[(size_t)(n0 + ln) * DFEAT + kb + lk];
        b.x = bp[0]; b.y = bp[1];
        acc2 = __builtin_amdgcn_wmma_f32_16x16x4_f32(
            false, a, false, b, (short)0, acc2, false, false);
    }

    // epilogue: per-row mean scale, add, ReLU, store.
    // D layout: lane n = n0+ln, m = j + 8*(lane>=16)
    const int mh = (lane >> 4) * 8;
#pragma unroll
    for (int j = 0; j < 8; ++j) {
        int row = m0 + mh + j;
        float v = fmaxf(scale[mh + j] * acc1[j] + acc2[j], 0.0f);
        if (row < n_nodes)
            out[(size_t)row * DFEAT + n0 + ln] = v;
    }
}

// ---------------------------------------------------------------------------
// Host launch
// ---------------------------------------------------------------------------
extern "C" void kernel_launch(void* const* d_in, const int* in_sizes, int n_in,
                              void* d_out, int out_size, void* d_ws, size_t ws_size,
                              hipStream_t stream) {
    const float* x  = (const float*)d_in[0];
    const int*   ei = (const int*)  d_in[1];   // edge_index, [2, E] flat
    const float* Wl = (const float*)d_in[2];
    const float* bl = (const float*)d_in[3];
    const float* Wr = (const float*)d_in[4];
    float* out = (float*)d_out;

    const int n_nodes = in_sizes[0] / DFEAT;   // 100000
    const int n_edges = in_sizes[1] / 2;       // 1600000

    float* msg = (float*)d_ws;                             // [N, 128]
    float* deg = msg + (size_t)n_nodes * DFEAT;            // [N]
    size_t zbytes = ((size_t)n_nodes * DFEAT + (size_t)n_nodes) * sizeof(float);
    (void)hipMemsetAsync(d_ws, 0, zbytes, stream);

    {   // scatter: one wave per edge
        long long threads = (long long)n_edges * 32;
        int blocks = (int)((threads + 255) / 256);
        sage_scatter_kernel<<<blocks, 256, 0, stream>>>(x, ei, msg, deg, n_edges);
    }
    {   // fused dual-GEMM + bias + relu
        int mtiles = (n_nodes + TM - 1) / TM;              // 6250, exact tiles
        sage_gemm_kernel<<<mtiles, 256, 0, stream>>>(x, msg, deg, Wl, bl, Wr,
                                                     out, n_nodes);
    }
}